// HolographicField_2018634629305
// MI455X (gfx1250) — compile-verified
//
#include <hip/hip_runtime.h>
#include <math.h>

// HolographicField for MI455X (gfx1250, wave32).
// Stages: A) per-row complex dot (mem-bound, 256MB read)
//         B) key correlations via V_WMMA_F32_16X16X4_F32 + exact LDS twiddle table
//         C) softmax(1024)
//         D) clean_key synthesis (LDS table FMAs)
//         E) broadcast complex multiply (mem-bound, 128MB read / 128MB write)

typedef __attribute__((ext_vector_type(2))) float v2f;
typedef __attribute__((ext_vector_type(8))) float v8f;

#define W_DIM 4096
#define S_DIM 4096
#define N_KEYS 1024
#define TWO_PI_F 6.283185307179586f
#define SHARP 50.0f

// ---------------------------------------------------------------- twiddle table
// tab[j] = (cos, sin)(j * 2pi/4096). Phases of all keys are exact multiples of
// 2pi/4096, so table lookup with index (w*(n+1)) & 4095 is exact.
__global__ void k_twiddle(float2* __restrict__ tab) {
    int j = blockIdx.x * blockDim.x + threadIdx.x;
    float s, c;
    __sincosf((float)j * (TWO_PI_F / 4096.0f), &s, &c);
    tab[j] = make_float2(c, s);
}

// ---------------------------------------------------------------- stage A
// mix[w] = sum_s holo[w,s] * conj(cue[w,s]).  One block per row, float4 loads.
__global__ void __launch_bounds__(256) k_rowdot(
        const float* __restrict__ hr, const float* __restrict__ hi,
        const float* __restrict__ cr, const float* __restrict__ ci,
        float2* __restrict__ mix) {
    __shared__ float2 red[256];
    const int w = blockIdx.x;
    const size_t off = (size_t)w * S_DIM;
    const float4* hr4 = (const float4*)(hr + off);
    const float4* hi4 = (const float4*)(hi + off);
    const float4* cr4 = (const float4*)(cr + off);
    const float4* ci4 = (const float4*)(ci + off);

    float re = 0.0f, im = 0.0f;
#pragma unroll
    for (int i = 0; i < 4; ++i) {
        int idx = i * 256 + threadIdx.x;
        float4 a = hr4[idx], b = hi4[idx], c = cr4[idx], d = ci4[idx];
        // holo*conj(cue): re = hr*cr + hi*ci ; im = hi*cr - hr*ci
        re = fmaf(a.x, c.x, re); re = fmaf(b.x, d.x, re);
        re = fmaf(a.y, c.y, re); re = fmaf(b.y, d.y, re);
        re = fmaf(a.z, c.z, re); re = fmaf(b.z, d.z, re);
        re = fmaf(a.w, c.w, re); re = fmaf(b.w, d.w, re);
        im = fmaf(b.x, c.x, im); im = fmaf(-a.x, d.x, im);
        im = fmaf(b.y, c.y, im); im = fmaf(-a.y, d.y, im);
        im = fmaf(b.z, c.z, im); im = fmaf(-a.z, d.z, im);
        im = fmaf(b.w, c.w, im); im = fmaf(-a.w, d.w, im);
    }
    red[threadIdx.x] = make_float2(re, im);
    __syncthreads();
    for (int s = 128; s > 0; s >>= 1) {
        if ((int)threadIdx.x < s) {
            red[threadIdx.x].x += red[threadIdx.x + s].x;
            red[threadIdx.x].y += red[threadIdx.x + s].y;
        }
        __syncthreads();
    }
    if (threadIdx.x == 0) mix[w] = red[0];
}

// ---------------------------------------------------------------- stage B (WMMA)
// corr_{re,im}[n] = (Re,Im) sum_w e^{i th(n,w)} * mix[w], as a real GEMM with
// K interleaved as (cos,sin) pairs:
//   A[m, 2w+0] = cos th, A[m, 2w+1] = sin th            (16 keys per wave tile)
//   B[2w+0, 0] =  mr,  B[2w+0, 1] = mi
//   B[2w+1, 0] = -mi,  B[2w+1, 1] = mr   (cols 2..15 = 0)
// -> C[m,0] = corr_re, C[m,1] = corr_im.  V_WMMA_F32_16X16X4_F32, f32-exact.
// 8 blocks x 8 waves; each wave owns 16 keys, K loop = 8192/4 = 2048 WMMAs.
__global__ void __launch_bounds__(256) k_corr_wmma(
        const float2* __restrict__ tab_g, const float2* __restrict__ mix_g,
        float* __restrict__ corr_re, float* __restrict__ corr_im) {
    __shared__ float2 s_tab[4096];
    for (int i = threadIdx.x; i < 4096; i += 256) s_tab[i] = tab_g[i];
    __syncthreads();

    const int lane = threadIdx.x & 31;
    const int wv   = threadIdx.x >> 5;
    const int key0 = blockIdx.x * 128 + wv * 16;
    const int m    = lane & 15;        // A row within tile / B column
    const int half = lane >> 4;        // VGPR0 carries K = 2*half (cos), VGPR1 K = 2*half+1 (sin)
    const int f    = key0 + m + 1;     // key frequency multiplier

    // B-column selectors, fixed per lane
    const float sel0 = (m == 0) ? 1.0f : 0.0f;
    const float sel1 = (m == 1) ? 1.0f : 0.0f;

    int idx  = (half * f) & 4095;      // twiddle index for w = half at k0 = 0
    const int step = (2 * f) & 4095;   // w advances by 2 each iteration
    int w = half;

    v8f acc = {};
    for (int k0 = 0; k0 < 2 * W_DIM; k0 += 4) {
        float2 t  = s_tab[idx];        // (cos, sin) for this lane's w
        float2 mx = mix_g[w];          // (mr, mi), uniform per half-wave (L2-hot)
        v2f a;  a.x = t.x;  a.y = t.y; // VGPR0 = cos term, VGPR1 = sin term
        v2f b;
        b.x = sel0 * mx.x + sel1 * mx.y;   // K even row:  [mr, mi, 0...]
        b.y = sel1 * mx.x - sel0 * mx.y;   // K odd  row:  [-mi, mr, 0...]
        acc = __builtin_amdgcn_wmma_f32_16x16x4_f32(
                false, a, false, b, (short)0, acc, false, false);
        idx = (idx + step) & 4095;
        w += 2;
    }

    // D 16x16 f32 layout: row = r + 8*(lane>=16), col = lane&15, vgpr r=0..7
    const int col = lane & 15;
    const int mo  = (lane >> 4) * 8;
    if (col < 2) {
        float* dst = (col == 0) ? corr_re : corr_im;
#pragma unroll
        for (int r = 0; r < 8; ++r) dst[key0 + mo + r] = acc[r];
    }
}

// ---------------------------------------------------------------- softmax
__global__ void __launch_bounds__(1024) k_softmax(
        const float* __restrict__ cr, const float* __restrict__ ci,
        float* __restrict__ attn) {
    __shared__ float red[1024];
    const int t = threadIdx.x;
    float c = sqrtf(cr[t] * cr[t] + ci[t] * ci[t]) * SHARP;
    red[t] = c;
    __syncthreads();
    for (int s = 512; s > 0; s >>= 1) {
        if (t < s) red[t] = fmaxf(red[t], red[t + s]);
        __syncthreads();
    }
    const float mx = red[0];
    __syncthreads();
    const float e = __expf(c - mx);
    red[t] = e;
    __syncthreads();
    for (int s = 512; s > 0; s >>= 1) {
        if (t < s) red[t] += red[t + s];
        __syncthreads();
    }
    attn[t] = e / red[0];
}

// ---------------------------------------------------------------- stage D
// clean[w] = sum_n attn[n] * twiddle[(w*(n+1)) & 4095]
__global__ void __launch_bounds__(256) k_clean(
        const float2* __restrict__ tab_g, const float* __restrict__ attn_g,
        float2* __restrict__ clean) {
    __shared__ float2 s_tab[4096];
    __shared__ float  s_attn[N_KEYS];
    for (int i = threadIdx.x; i < 4096; i += 256) s_tab[i] = tab_g[i];
    for (int i = threadIdx.x; i < N_KEYS; i += 256) s_attn[i] = attn_g[i];
    __syncthreads();

    const int w = blockIdx.x * 256 + threadIdx.x;
    int idx = w & 4095;                 // phase index for n = 0 (f = 1)
    float re = 0.0f, im = 0.0f;
    for (int n = 0; n < N_KEYS; ++n) {
        float a = s_attn[n];
        float2 t = s_tab[idx];
        re = fmaf(a, t.x, re);
        im = fmaf(a, t.y, im);
        idx = (idx + w) & 4095;         // f advances by 1 -> index advances by w
    }
    clean[w] = make_float2(re, im);
}

// ---------------------------------------------------------------- stage E
// out[0] = Re(holo*clean[w]), out[1] = Im(holo*clean[w]); float4 streaming.
__global__ void __launch_bounds__(256) k_apply(
        const float4* __restrict__ hr4, const float4* __restrict__ hi4,
        const float2* __restrict__ clean,
        float4* __restrict__ out_re, float4* __restrict__ out_im) {
    const size_t g = (size_t)blockIdx.x * 256 + threadIdx.x;
    const int w = (int)(g >> 10);       // S/4 = 1024 float4 per row (uniform per block)
    const float2 k = clean[w];
    float4 a = hr4[g], b = hi4[g];
    float4 orr, oim;
    orr.x = a.x * k.x - b.x * k.y;  oim.x = a.x * k.y + b.x * k.x;
    orr.y = a.y * k.x - b.y * k.y;  oim.y = a.y * k.y + b.y * k.x;
    orr.z = a.z * k.x - b.z * k.y;  oim.z = a.z * k.y + b.z * k.x;
    orr.w = a.w * k.x - b.w * k.y;  oim.w = a.w * k.y + b.w * k.x;
    out_re[g] = orr;
    out_im[g] = oim;
}

// ---------------------------------------------------------------- launch
extern "C" void kernel_launch(void* const* d_in, const int* in_sizes, int n_in,
                              void* d_out, int out_size, void* d_ws, size_t ws_size,
                              hipStream_t stream) {
    (void)in_sizes; (void)n_in; (void)out_size; (void)ws_size;
    const float* hr = (const float*)d_in[0];
    const float* hi = (const float*)d_in[1];
    const float* cr = (const float*)d_in[2];
    const float* ci = (const float*)d_in[3];
    // d_in[4] = n_keys (device int); reference fixes it at 1024.

    float*  ws      = (float*)d_ws;
    float2* tab     = (float2*)(ws);            // 4096 float2 (32 KB)
    float2* mix     = (float2*)(ws + 8192);     // 4096 float2 (32 KB)
    float*  corr_re = ws + 16384;               // 1024 f32
    float*  corr_im = ws + 17408;               // 1024 f32
    float*  attn    = ws + 18432;               // 1024 f32
    float2* clean   = (float2*)(ws + 19456);    // 4096 float2 (32 KB)

    float* out = (float*)d_out;                 // [2, 4096, 4096]

    k_twiddle  <<<W_DIM / 256, 256, 0, stream>>>(tab);
    k_rowdot   <<<W_DIM,       256, 0, stream>>>(hr, hi, cr, ci, mix);
    k_corr_wmma<<<N_KEYS / 128, 256, 0, stream>>>(tab, mix, corr_re, corr_im);
    k_softmax  <<<1,          1024, 0, stream>>>(corr_re, corr_im, attn);
    k_clean    <<<W_DIM / 256, 256, 0, stream>>>(tab, attn, clean);

    const size_t nvec4 = (size_t)W_DIM * S_DIM / 4;   // 4,194,304 float4 per plane
    k_apply<<<(unsigned)(nvec4 / 256), 256, 0, stream>>>(
        (const float4*)hr, (const float4*)hi, clean,
        (float4*)out, (float4*)out + nvec4);
}